// Sampler_34127810134265
// MI455X (gfx1250) — compile-verified
//
#include <hip/hip_runtime.h>
#include <hip/hip_bf16.h>
#include <stdint.h>

#define B_ROWS 128
#define V_SIZE 128000
#define CHUNKS 4
#define CHUNK (V_SIZE / CHUNKS)   // 32000
#define N4 (CHUNK / 4)            // 8000 float4 per chunk
#define NBINS 4096
#define BIN_SHIFT 20
#define CAP 2048
#define TOPK 50
#define TOPP 0.9f
#define INVT (1.0f / 0.7f)

typedef __attribute__((ext_vector_type(2))) float v2f;
typedef __attribute__((ext_vector_type(8))) float v8f;

// Monotone float -> uint mapping (ascending float == ascending uint)
__device__ __forceinline__ unsigned okey(float x) {
  unsigned u = __float_as_uint(x);
  return (u & 0x80000000u) ? ~u : (u | 0x80000000u);
}
__device__ __forceinline__ float okey_inv(unsigned k) {
  unsigned u = (k & 0x80000000u) ? (k & 0x7fffffffu) : ~k;
  return __uint_as_float(u);
}

// Wave32 sum reduction via V_WMMA_F32_16X16X4_F32 with B = ones.
// A layout (ISA 7.12.2): VGPR0 lanes0-15 -> A[lane][0], lanes16-31 -> A[lane-16][2].
// With a.y = 0: D[m][n] = s_m + s_{m+16}. Summing the 8 D VGPRs per lane gives
// rows 0..7 (lanes<16) / rows 8..15 (lanes>=16); one xor-16 shuffle completes the sum.
__device__ __forceinline__ float wave_sum_wmma(float s) {
  v2f a;    a.x = s;    a.y = 0.0f;
  v2f ones; ones.x = 1.0f; ones.y = 1.0f;
  v8f c = {};
  c = __builtin_amdgcn_wmma_f32_16x16x4_f32(false, a, false, ones, (short)0, c,
                                            false, false);
  float t = c[0] + c[1] + c[2] + c[3] + c[4] + c[5] + c[6] + c[7];
  t += __shfl_xor(t, 16, 32);
  return t;
}

__global__ void k_zero(unsigned* ws, int n) {
  int i = blockIdx.x * blockDim.x + threadIdx.x;
  int stride = gridDim.x * blockDim.x;
  for (; i < n; i += stride) ws[i] = 0u;
}

// Pass 1: per (row, chunk): chunk max, chunk sum(exp(x-max)), per-row histogram.
__global__ void k_stats(const float* __restrict__ logits, unsigned* __restrict__ hist,
                        float* __restrict__ cmax, float* __restrict__ csum) {
  const int chunk = blockIdx.x, row = blockIdx.y, tid = threadIdx.x; // 256 threads
  __shared__ unsigned lh[NBINS];
  __shared__ float red[256];
  __shared__ float wred[8];
  for (int i = tid; i < NBINS; i += 256) lh[i] = 0u;
  __syncthreads();

  const float4* p = (const float4*)(logits + (size_t)row * V_SIZE + (size_t)chunk * CHUNK);
  float m = -3.4e38f;
  for (int i = tid; i < N4; i += 256) {
    if (i + 512 < N4) __builtin_prefetch(p + i + 512, 0, 1); // global_prefetch_b8
    float4 v = p[i];
    float x0 = v.x * INVT, x1 = v.y * INVT, x2 = v.z * INVT, x3 = v.w * INVT;
    atomicAdd(&lh[okey(x0) >> BIN_SHIFT], 1u);
    atomicAdd(&lh[okey(x1) >> BIN_SHIFT], 1u);
    atomicAdd(&lh[okey(x2) >> BIN_SHIFT], 1u);
    atomicAdd(&lh[okey(x3) >> BIN_SHIFT], 1u);
    m = fmaxf(m, fmaxf(fmaxf(x0, x1), fmaxf(x2, x3)));
  }
  red[tid] = m;
  __syncthreads();
  for (int off = 128; off > 0; off >>= 1) {
    if (tid < off) red[tid] = fmaxf(red[tid], red[tid + off]);
    __syncthreads();
  }
  const float mc = red[0];

  float s = 0.f;  // second pass over chunk: L2-resident
  for (int i = tid; i < N4; i += 256) {
    float4 v = p[i];
    s += expf(v.x * INVT - mc) + expf(v.y * INVT - mc) +
         expf(v.z * INVT - mc) + expf(v.w * INVT - mc);
  }
  float wsum = wave_sum_wmma(s);          // v_wmma_f32_16x16x4_f32
  const int lane = tid & 31, wave = tid >> 5;
  if (lane == 0) wred[wave] = wsum;
  __syncthreads();
  if (tid == 0) {
    float tot = 0.f;
    for (int w = 0; w < 8; ++w) tot += wred[w];
    cmax[row * CHUNKS + chunk] = mc;
    csum[row * CHUNKS + chunk] = tot;
  }
  for (int i = tid; i < NBINS; i += 256) {
    unsigned c = lh[i];
    if (c) atomicAdd(&hist[(size_t)row * NBINS + i], c);
  }
}

// Pass 2: per row: combine chunk stats -> M, Z; scan histogram from the top
// to find the bin threshold covering the top-K values.
__global__ void k_combine(const unsigned* __restrict__ hist,
                          const float* __restrict__ cmax, const float* __restrict__ csum,
                          float* __restrict__ rowM, float* __restrict__ rowZ,
                          unsigned* __restrict__ rowTau) {
  const int row = blockIdx.x, tid = threadIdx.x; // 128 threads
  __shared__ unsigned lh[NBINS];
  __shared__ unsigned seg[128];
  for (int i = tid; i < NBINS; i += 128) lh[i] = hist[(size_t)row * NBINS + i];
  __syncthreads();
  // segment t = 32 bins descending from the top
  unsigned ssum = 0;
  const int hi = NBINS - 1 - 32 * tid;
  for (int b = hi; b > hi - 32; --b) ssum += lh[b];
  seg[tid] = ssum;
  __syncthreads();
  if (tid == 0) {
    float M = cmax[row * CHUNKS];
    for (int c = 1; c < CHUNKS; ++c) M = fmaxf(M, cmax[row * CHUNKS + c]);
    float Z = 0.f;
    for (int c = 0; c < CHUNKS; ++c) Z += csum[row * CHUNKS + c] * expf(cmax[row * CHUNKS + c] - M);
    rowM[row] = M;
    rowZ[row] = Z;
    unsigned cum = 0;
    int s = 0;
    while (s < 128 && cum + seg[s] < TOPK) { cum += seg[s]; ++s; }
    int tb = 0;
    if (s < 128) {
      const int top = NBINS - 1 - 32 * s;
      tb = top - 31;
      for (int b = top; b >= top - 31; --b) {
        cum += lh[b];
        if (cum >= TOPK) { tb = b; break; }
      }
    }
    rowTau[row] = ((unsigned)tb) << BIN_SHIFT;
  }
}

// Pass 3: gather all candidates with key >= tau (superset of top-K). L2-hot re-read.
__global__ void k_collect(const float* __restrict__ logits, const unsigned* __restrict__ rowTau,
                          unsigned* __restrict__ cnt, uint2* __restrict__ cand) {
  const int chunk = blockIdx.x, row = blockIdx.y, tid = threadIdx.x; // 256 threads
  const unsigned tau = rowTau[row];
  const float4* p = (const float4*)(logits + (size_t)row * V_SIZE + (size_t)chunk * CHUNK);
  for (int i = tid; i < N4; i += 256) {
    float4 v = p[i];
    const float xs[4] = {v.x, v.y, v.z, v.w};
#pragma unroll
    for (int c = 0; c < 4; ++c) {
      const unsigned k = okey(xs[c] * INVT);
      if (k >= tau) {
        unsigned pos = atomicAdd(&cnt[row], 1u);
        if (pos < CAP)
          cand[(size_t)row * CAP + pos] =
              make_uint2(k, (unsigned)(chunk * CHUNK + 4 * i + c));
      }
    }
  }
}

// Pass 4: per row: bitonic sort candidates descending (ties: index descending,
// matching reversed stable ascending argsort), apply top-p prefix rule + top-k,
// renormalize, inverse-CDF sample in original index order.
__global__ void k_final(const uint2* __restrict__ cand, const unsigned* __restrict__ cnt,
                        const float* __restrict__ rowM, const float* __restrict__ rowZ,
                        const float* __restrict__ uarr, float* __restrict__ out) {
  const int row = blockIdx.x, tid = threadIdx.x; // 256 threads
  __shared__ unsigned long long sk[CAP];
  __shared__ float es[64];
  __shared__ unsigned si[64];
  const int n = (int)min(cnt[row], (unsigned)CAP);
  for (int i = tid; i < CAP; i += 256) {
    if (i < n) {
      uint2 c = cand[(size_t)row * CAP + i];
      sk[i] = ((unsigned long long)c.x << 32) | (unsigned long long)c.y;
    } else {
      sk[i] = 0ull;
    }
  }
  __syncthreads();
  // Bitonic sort, descending on (key, idx)
  for (unsigned k = 2; k <= CAP; k <<= 1) {
    for (unsigned j = k >> 1; j > 0; j >>= 1) {
      for (unsigned i = tid; i < CAP; i += 256) {
        const unsigned ixj = i ^ j;
        if (ixj > i) {
          unsigned long long a = sk[i], b = sk[ixj];
          const bool desc = ((i & k) == 0);
          if (desc ? (a < b) : (a > b)) { sk[i] = b; sk[ixj] = a; }
        }
      }
      __syncthreads();
    }
  }
  const float M = rowM[row], Z = rowZ[row];
  if (tid < 64) {
    if (tid < n) {
      const unsigned long long e = sk[tid];
      es[tid] = expf(okey_inv((unsigned)(e >> 32)) - M);
      si[tid] = (unsigned)(e & 0xffffffffu);
    } else {
      es[tid] = 0.f;
      si[tid] = 0u;
    }
  }
  __syncthreads();
  if (tid == 0) {
    const int kmax = n < TOPK ? n : TOPK;
    // token i kept iff sum of probs strictly above it < top_p  <=>  cumBefore < top_p * Z
    float cb = 0.f;
    int L = 0;
    for (int i = 0; i < kmax; ++i) {
      if (i > 0 && cb >= TOPP * Z) break;
      cb += es[i];
      ++L;
    }
    const float S = cb;  // kept mass (unnormalized)
    // order kept tokens by original index ascending (L <= 50)
    for (int i = 0; i < L - 1; ++i) {
      int mi = i;
      for (int j = i + 1; j < L; ++j)
        if (si[j] < si[mi]) mi = j;
      if (mi != i) {
        unsigned ti = si[i]; si[i] = si[mi]; si[mi] = ti;
        float te = es[i];    es[i] = es[mi]; es[mi] = te;
      }
    }
    const float target = uarr[row] * S;
    float cum = 0.f;
    unsigned tok = (L > 0) ? si[L - 1] : 0u;
    for (int i = 0; i < L; ++i) {
      cum += es[i];
      if (cum >= target) { tok = si[i]; break; }
    }
    out[row] = (float)tok;
  }
}

extern "C" void kernel_launch(void* const* d_in, const int* in_sizes, int n_in,
                              void* d_out, int out_size, void* d_ws, size_t ws_size,
                              hipStream_t stream) {
  (void)in_sizes; (void)n_in; (void)out_size; (void)ws_size;
  const float* logits = (const float*)d_in[0];
  // d_in[1] = input_ids (unused: repetition_penalty == 1.0)
  const float* u = (const float*)d_in[2];
  float* out = (float*)d_out;

  // workspace layout (~4.1 MB)
  unsigned* hist = (unsigned*)d_ws;                       // B*NBINS
  unsigned* cnt  = hist + (size_t)B_ROWS * NBINS;         // B
  float* cmax = (float*)(cnt + B_ROWS);                   // B*CHUNKS
  float* csum = cmax + B_ROWS * CHUNKS;                   // B*CHUNKS
  float* rowM = csum + B_ROWS * CHUNKS;                   // B
  float* rowZ = rowM + B_ROWS;                            // B
  unsigned* rowTau = (unsigned*)(rowZ + B_ROWS);          // B
  uint2* cand = (uint2*)((((uintptr_t)(rowTau + B_ROWS)) + 15) & ~(uintptr_t)15);

  k_zero<<<256, 256, 0, stream>>>(hist, B_ROWS * NBINS + B_ROWS);
  dim3 gs(CHUNKS, B_ROWS);
  k_stats<<<gs, 256, 0, stream>>>(logits, hist, cmax, csum);
  k_combine<<<B_ROWS, 128, 0, stream>>>(hist, cmax, csum, rowM, rowZ, rowTau);
  k_collect<<<gs, 256, 0, stream>>>(logits, rowTau, cnt, cand);
  k_final<<<B_ROWS, 256, 0, stream>>>(cand, cnt, rowM, rowZ, u, out);
}